// GCN_69097433858404
// MI455X (gfx1250) — compile-verified
//
#include <hip/hip_runtime.h>
#include <math.h>

typedef __attribute__((ext_vector_type(2))) float v2f;
typedef __attribute__((ext_vector_type(8))) float v8f;

#define TB 256

// ---------------- degree / normalization ----------------

__global__ void k_init_deg(float* __restrict__ deg, int n) {
    int i = blockIdx.x * blockDim.x + threadIdx.x;
    if (i < n) deg[i] = 1.0f;                       // self-loop contribution
}

__global__ void k_accum_deg(const long long* __restrict__ cols,
                            float* __restrict__ deg, int nE) {
    int e = blockIdx.x * blockDim.x + threadIdx.x;
    if (e < nE) unsafeAtomicAdd(&deg[cols[e]], 1.0f);   // exact integer counts in f32
}

__global__ void k_dinv(float* __restrict__ d, int n) {
    int i = blockIdx.x * blockDim.x + threadIdx.x;
    if (i < n) {
        float v = d[i];
        d[i] = (v > 0.0f) ? (1.0f / sqrtf(v)) : 0.0f;
    }
}

// ---------------- WMMA f32 GEMM: C[M x N] = op(A)[M x K] * B[K x N] ----------------
// One 16x16 output tile per wave32. V_WMMA_F32_16X16X4_F32, K stepped by 4.
// A frag (16x4 f32): lane<16 -> row=l, {K, K+1}; lane>=16 -> row=l-16, {K+2, K+3}
// B frag (4x16 f32): mirrored (lane<16 -> col=l, rows K/K+1; hi half rows K+2/K+3)
// C/D (16x16 f32, 8 VGPRs): vgpr j, lane<16 -> M=j, N=l; lane>=16 -> M=j+8, N=l-16

template <int N, int K, bool RELU_A>
__global__ __launch_bounds__(128) void k_gemm_wmma(
        const float* __restrict__ A, const float* __restrict__ B,
        float* __restrict__ C, int M) {
    const int lane = threadIdx.x & 31;
    const int wave = threadIdx.x >> 5;
    const int nT   = N / 16;
    const int mT   = (M + 15) >> 4;
    const int t    = blockIdx.x * 4 + wave;       // wave-uniform
    if (t >= mT * nT) return;                     // uniform exit: EXEC stays full

    const int mTile = t / nT;
    const int nTile = t - mTile * nT;
    const int l16   = lane & 15;
    const int hi    = lane >> 4;                  // 0 or 1
    const int kh    = hi * 2;                     // K sub-offset 0 or 2
    int arow = mTile * 16 + l16;
    if (arow >= M) arow = M - 1;                  // clamp for ragged M (unused here)
    const int col = nTile * 16 + l16;

    const float* aBase = A + (size_t)arow * K + kh;
    const float* bBase = B + (size_t)kh * N + col;

    v8f acc = {0.f, 0.f, 0.f, 0.f, 0.f, 0.f, 0.f, 0.f};
#pragma unroll
    for (int k = 0; k < K; k += 4) {
        float2 av = *(const float2*)(aBase + k);  // b64 load, 8B aligned (kh+k even)
        v2f a;
        a.x = RELU_A ? fmaxf(av.x, 0.0f) : av.x;
        a.y = RELU_A ? fmaxf(av.y, 0.0f) : av.y;
        v2f b;
        b.x = bBase[(size_t)k * N];
        b.y = bBase[(size_t)(k + 1) * N];
        acc = __builtin_amdgcn_wmma_f32_16x16x4_f32(
                  false, a, false, b, (short)0, acc, false, false);
    }

    const int crow0 = mTile * 16 + hi * 8;
#pragma unroll
    for (int j = 0; j < 8; ++j) {
        int r = crow0 + j;
        if (r < M) C[(size_t)r * N + col] = acc[j];
    }
}

// ---------------- self-loop init (+bias): agg[i] = hw[i]*dinv[i]^2 + b ----------------

template <int F>
__global__ void k_init_agg(const float* __restrict__ hw, const float* __restrict__ dinv,
                           const float* __restrict__ bias, float* __restrict__ agg, int n) {
    const int chunks = F / 4;
    long long tid = (long long)blockIdx.x * blockDim.x + threadIdx.x;
    long long i = tid / chunks;
    int c = (int)(tid % chunks) * 4;
    if (i >= n) return;
    float d  = dinv[i];
    float d2 = d * d;
    float4 v = *(const float4*)(hw + i * F + c);
    float4 o;
    o.x = v.x * d2 + bias[c + 0];
    o.y = v.y * d2 + bias[c + 1];
    o.z = v.z * d2 + bias[c + 2];
    o.w = v.w * d2 + bias[c + 3];
    *(float4*)(agg + i * F + c) = o;
}

// ---------------- edge scatter: agg[col] += hw[row] * dinv[row]*dinv[col] ----------------

template <int F>
__global__ void k_scatter(const float* __restrict__ hw,
                          const long long* __restrict__ rows,
                          const long long* __restrict__ cols,
                          const float* __restrict__ dinv,
                          float* __restrict__ agg, int nE) {
    const int chunks = F / 4;
    long long tid = (long long)blockIdx.x * blockDim.x + threadIdx.x;
    long long e = tid / chunks;
    int c = (int)(tid % chunks) * 4;
    if (e >= nE) return;
    const long long r = rows[e];
    const long long d = cols[e];
    const float norm = dinv[r] * dinv[d];
    float4 v = *(const float4*)(hw + r * (long long)F + c);   // b128 gather
    float* dst = agg + d * (long long)F + c;
    unsafeAtomicAdd(dst + 0, v.x * norm);   // global_atomic_add_f32
    unsafeAtomicAdd(dst + 1, v.y * norm);
    unsafeAtomicAdd(dst + 2, v.z * norm);
    unsafeAtomicAdd(dst + 3, v.w * norm);
}

// ---------------- launch ----------------

extern "C" void kernel_launch(void* const* d_in, const int* in_sizes, int n_in,
                              void* d_out, int out_size, void* d_ws, size_t ws_size,
                              hipStream_t stream) {
    const float*      x    = (const float*)d_in[0];
    const long long*  ei   = (const long long*)d_in[1];   // int64 edge_index [2, E]
    const float*      W1   = (const float*)d_in[3];
    const float*      b1   = (const float*)d_in[4];
    const float*      W2   = (const float*)d_in[5];
    const float*      b2   = (const float*)d_in[6];

    const int nNodes = in_sizes[0] / 128;
    const int nEdges = in_sizes[2];                // edge_weight length == E
    const long long* rows = ei;
    const long long* cols = ei + nEdges;

    char* ws = (char*)d_ws;
    size_t off = 0;
    float* dinv = (float*)(ws + off); off += (((size_t)nNodes * 4) + 255) & ~(size_t)255;
    float* hw   = (float*)(ws + off); off += (((size_t)nNodes * 128 * 4) + 255) & ~(size_t)255;
    float* agg1 = (float*)(ws + off);              // 51.2 MB; total ws use ~103 MB
    float* out  = (float*)d_out;

    // --- degrees -> dinv (shared by both layers) ---
    k_init_deg <<<(nNodes + TB - 1) / TB, TB, 0, stream>>>(dinv, nNodes);
    k_accum_deg<<<(nEdges + TB - 1) / TB, TB, 0, stream>>>(cols, dinv, nEdges);
    k_dinv     <<<(nNodes + TB - 1) / TB, TB, 0, stream>>>(dinv, nNodes);

    // --- layer 1: hw = x @ W1 ; agg1 = scatter(hw*norm) + b1 ---
    {
        const int tiles = ((nNodes + 15) / 16) * (128 / 16);
        k_gemm_wmma<128, 128, false><<<(tiles + 3) / 4, 128, 0, stream>>>(x, W1, hw, nNodes);
        long long itot = (long long)nNodes * 32;
        k_init_agg<128><<<(int)((itot + TB - 1) / TB), TB, 0, stream>>>(hw, dinv, b1, agg1, nNodes);
        long long stot = (long long)nEdges * 32;
        k_scatter<128><<<(int)((stot + TB - 1) / TB), TB, 0, stream>>>(hw, rows, cols, dinv, agg1, nEdges);
    }

    // --- layer 2: hw2 = relu(agg1) @ W2 (relu fused into A load); out = scatter + b2 ---
    {
        const int tiles = ((nNodes + 15) / 16) * (64 / 16);
        k_gemm_wmma<64, 128, true><<<(tiles + 3) / 4, 128, 0, stream>>>(agg1, W2, hw, nNodes);
        long long itot = (long long)nNodes * 16;
        k_init_agg<64><<<(int)((itot + TB - 1) / TB), TB, 0, stream>>>(hw, dinv, b2, out, nNodes);
        long long stot = (long long)nEdges * 16;
        k_scatter<64><<<(int)((stot + TB - 1) / TB), TB, 0, stream>>>(hw, rows, cols, dinv, out, nEdges);
    }
}